// Up_36386962932319
// MI455X (gfx1250) — compile-verified
//
#include <hip/hip_runtime.h>
#include <hip/hip_bf16.h>

typedef __attribute__((ext_vector_type(2))) float v2f;
typedef __attribute__((ext_vector_type(8))) float v8f;

#define BATCH   4
#define HH      64
#define WW      64
#define LSEQ    4096
#define MTOK    16384   // BATCH*LSEQ
#define DA      384
#define DB      128

__device__ __forceinline__ float silu_f(float x) { return x / (1.0f + expf(-x)); }
__device__ __forceinline__ float softplus_f(float x) { return x > 20.0f ? x : log1pf(expf(x)); }

// ---------------------------------------------------------------------------
// 1) bilinear up2(x1) + concat(x2) -> t[b][l][c], c fastest, C=384
// ---------------------------------------------------------------------------
__global__ __launch_bounds__(256) void prep_kernel(const float* __restrict__ x1,
                                                   const float* __restrict__ x2,
                                                   float* __restrict__ t) {
    int idx = blockIdx.x * 256 + threadIdx.x;          // exact: MTOK*384 threads
    int c  = idx % DA;
    int bl = idx / DA;
    int l  = bl & (LSEQ - 1);
    int b  = bl >> 12;
    int yy = l >> 6, xx = l & 63;
    float val;
    if (c < 128) {
        val = x2[(((size_t)b * 128 + c) * 64 + yy) * 64 + xx];
    } else {
        int cu = c - 128;
        const float* p = x1 + ((size_t)b * 256 + cu) * 32 * 32;
        const float sc = 31.0f / 63.0f;
        float ty = yy * sc; int y0 = (int)ty; int y1 = y0 + 1; if (y1 > 31) y1 = 31;
        float fy = ty - (float)y0;
        float tx = xx * sc; int x0 = (int)tx; int x1i = x0 + 1; if (x1i > 31) x1i = 31;
        float fx = tx - (float)x0;
        float r0 = p[y0 * 32 + x0]  * (1.0f - fy) + p[y1 * 32 + x0]  * fy;
        float r1 = p[y0 * 32 + x1i] * (1.0f - fy) + p[y1 * 32 + x1i] * fy;
        val = r0 * (1.0f - fx) + r1 * fx;
    }
    t[idx] = val;
}

// ---------------------------------------------------------------------------
// 2) GEMM via V_WMMA_F32_16X16X4_F32:  out[M,N] = act(A[M,K] @ W[N,K]^T + b)
//    256 threads = 8 waves; each wave owns one 16x16 tile of D.
//    f32 WMMA fragment layout (ISA 7.12.2): lane = half*16+lm
//      A frag: a[v] = A[m0+lm][k + 2*half + v]   (contiguous b64 per lane)
//      B frag: b[v] = W[n0+lm][k + 2*half + v]
//      D frag: d[v] = D[m0 + v + 8*half][n0+lm]
// ---------------------------------------------------------------------------
template <bool BIAS, bool SILU>
__global__ __launch_bounds__(256) void gemm_wmma(const float* __restrict__ A,
                                                 const float* __restrict__ W,
                                                 const float* __restrict__ bias,
                                                 float* __restrict__ out,
                                                 int M, int N, int K) {
    int lane = threadIdx.x & 31;
    int wave = threadIdx.x >> 5;
    int lm   = lane & 15;
    int half = lane >> 4;
    int n0 = (blockIdx.x * 8 + wave) * 16;
    int m0 = blockIdx.y * 16;

    const float* arow = A + (size_t)(m0 + lm) * K + 2 * half;
    const float* brow = W + (size_t)(n0 + lm) * K + 2 * half;

    v8f acc = {};
    for (int k = 0; k < K; k += 4) {
        v2f av = *(const v2f*)(arow + k);
        v2f bv = *(const v2f*)(brow + k);
        acc = __builtin_amdgcn_wmma_f32_16x16x4_f32(false, av, false, bv,
                                                    (short)0, acc, false, false);
    }

    float bval = 0.0f;
    if (BIAS) bval = bias[n0 + lm];
    float* orow = out + (size_t)m0 * N + n0 + lm;
#pragma unroll
    for (int v = 0; v < 8; ++v) {
        float x = acc[v] + bval;
        if (SILU) x = silu_f(x);
        orow[(size_t)(v + 8 * half) * N] = x;
    }
}

// ---------------------------------------------------------------------------
// 3) causal depthwise conv (DCONV=4) + SiLU:  xz[.., :d] -> xc
// ---------------------------------------------------------------------------
__global__ __launch_bounds__(256) void conv_kernel(const float* __restrict__ xz,
                                                   const float* __restrict__ cw,
                                                   const float* __restrict__ cb,
                                                   float* __restrict__ xc,
                                                   int d, int d2, int total) {
    int idx = blockIdx.x * 256 + threadIdx.x;
    if (idx >= total) return;
    int c  = idx % d;
    int bl = idx / d;
    int l  = bl & (LSEQ - 1);
    size_t rowbase = (size_t)(bl - l) * d2 + c;    // start of this batch
    float acc = cb[c];
#pragma unroll
    for (int k = 0; k < 4; ++k) {
        int li = l - 3 + k;
        if (li >= 0) acc += xz[rowbase + (size_t)li * d2] * cw[c * 4 + k];
    }
    xc[(size_t)bl * d + c] = silu_f(acc);
}

// ---------------------------------------------------------------------------
// 4) x_proj + dt_proj + softplus per token (1 block / token, 128 thr)
//    dbc[0:rank]=dt_in, [rank:rank+2]=Bm, [rank+2:rank+4]=Cm
// ---------------------------------------------------------------------------
__global__ __launch_bounds__(128) void xproj_kernel(const float* __restrict__ xc,
                                                    const float* __restrict__ xpw,
                                                    const float* __restrict__ dtw,
                                                    const float* __restrict__ dtb,
                                                    float* __restrict__ dt,
                                                    float* __restrict__ bc,
                                                    int d, int rank) {
    __shared__ float s_x[DA];
    __shared__ float s_dbc[32];
    int m   = blockIdx.x;
    int tid = threadIdx.x;
    int ndbc = rank + 4;
    for (int i = tid; i < d; i += 128) s_x[i] = xc[(size_t)m * d + i];
    __syncthreads();
    if (tid < ndbc) {
        const float* wr = xpw + (size_t)tid * d;
        float s = 0.0f;
        for (int j = 0; j < d; ++j) s += s_x[j] * wr[j];
        s_dbc[tid] = s;
    }
    __syncthreads();
    if (tid < 4) bc[(size_t)m * 4 + tid] = s_dbc[rank + tid];
    for (int o = tid; o < d; o += 128) {
        float s = dtb[o];
        const float* wr = dtw + (size_t)o * rank;
        for (int j = 0; j < rank; ++j) s += s_dbc[j] * wr[j];
        dt[(size_t)m * d + o] = softplus_f(s);
    }
}

// ---------------------------------------------------------------------------
// 5) selective scan: one thread per (b, c) recurrence, L steps, D_STATE=2
//    y = (h.C + xc*D) * silu(z)
// ---------------------------------------------------------------------------
__global__ __launch_bounds__(256) void scan_kernel(const float* __restrict__ xc,
                                                   const float* __restrict__ dt,
                                                   const float* __restrict__ bc,
                                                   const float* __restrict__ xz,  // z at +d
                                                   const float* __restrict__ Alog,
                                                   const float* __restrict__ Dp,
                                                   float* __restrict__ yout,
                                                   int d, int d2, int nthreads) {
    int tid = blockIdx.x * 256 + threadIdx.x;
    if (tid >= nthreads) return;
    int c = tid % d;
    int b = tid / d;
    float A0 = -expf(Alog[c * 2 + 0]);
    float A1 = -expf(Alog[c * 2 + 1]);
    float Dv = Dp[c];
    float h0 = 0.0f, h1 = 0.0f;
    size_t ix = (size_t)b * LSEQ * d + c;
    size_t iz = (size_t)b * LSEQ * d2 + d + c;
    size_t ib = (size_t)b * LSEQ * 4;
    for (int l = 0; l < LSEQ; ++l) {
        float dtv = dt[ix];
        float xv  = xc[ix];
        float B0 = bc[ib + 0], B1 = bc[ib + 1];
        float C0 = bc[ib + 2], C1 = bc[ib + 3];
        h0 = expf(dtv * A0) * h0 + dtv * B0 * xv;
        h1 = expf(dtv * A1) * h1 + dtv * B1 * xv;
        float y = h0 * C0 + h1 * C1 + xv * Dv;
        float zv = xz[iz];
        y *= silu_f(zv);
        yout[ix] = y;
        ix += d; iz += d2; ib += 4;
    }
}

// ---------------------------------------------------------------------------
// 6) LayerNorm, one wave per token
// ---------------------------------------------------------------------------
__global__ __launch_bounds__(256) void ln_kernel(const float* __restrict__ y,
                                                 const float* __restrict__ g,
                                                 const float* __restrict__ bt,
                                                 float* __restrict__ out,
                                                 int d, int M) {
    int gt = blockIdx.x * 256 + threadIdx.x;
    int wv = gt >> 5, lane = gt & 31;
    if (wv >= M) return;
    const float* row = y + (size_t)wv * d;
    float s = 0.0f, sq = 0.0f;
    for (int i = lane; i < d; i += 32) { float v = row[i]; s += v; sq += v * v; }
    for (int o = 16; o > 0; o >>= 1) { s += __shfl_xor(s, o, 32); sq += __shfl_xor(sq, o, 32); }
    float mu = s / d;
    float var = sq / d - mu * mu;
    float r = rsqrtf(var + 1e-5f);
    for (int i = lane; i < d; i += 32)
        out[(size_t)wv * d + i] = (row[i] - mu) * r * g[i] + bt[i];
}

// ---------------------------------------------------------------------------
// 7) final: LN_b(y_b) + br1, store NCHW
// ---------------------------------------------------------------------------
__global__ __launch_bounds__(256) void final_kernel(const float* __restrict__ yb,
                                                    const float* __restrict__ g,
                                                    const float* __restrict__ bt,
                                                    const float* __restrict__ br1,
                                                    float* __restrict__ out) {
    int gt = blockIdx.x * 256 + threadIdx.x;
    int wv = gt >> 5, lane = gt & 31;
    if (wv >= MTOK) return;
    const float* row = yb + (size_t)wv * DB;
    float s = 0.0f, sq = 0.0f;
    for (int i = lane; i < DB; i += 32) { float v = row[i]; s += v; sq += v * v; }
    for (int o = 16; o > 0; o >>= 1) { s += __shfl_xor(s, o, 32); sq += __shfl_xor(sq, o, 32); }
    float mu = s / DB;
    float var = sq / DB - mu * mu;
    float r = rsqrtf(var + 1e-5f);
    int b = wv >> 12, l = wv & (LSEQ - 1);
    int yy = l >> 6, xx = l & 63;
    for (int i = lane; i < DB; i += 32) {
        float v = (row[i] - mu) * r * g[i] + bt[i] + br1[(size_t)wv * DB + i];
        out[(((size_t)b * DB + i) * 64 + yy) * 64 + xx] = v;
    }
}

// ---------------------------------------------------------------------------
// launch
// ---------------------------------------------------------------------------
extern "C" void kernel_launch(void* const* d_in, const int* in_sizes, int n_in,
                              void* d_out, int out_size, void* d_ws, size_t ws_size,
                              hipStream_t stream) {
    const float* x1      = (const float*)d_in[0];
    const float* x2      = (const float*)d_in[1];
    const float* lin1_w  = (const float*)d_in[2];
    const float* lin1_b  = (const float*)d_in[3];
    const float* a_inw   = (const float*)d_in[4];
    const float* a_cw    = (const float*)d_in[5];
    const float* a_cb    = (const float*)d_in[6];
    const float* a_xpw   = (const float*)d_in[7];
    const float* a_dtw   = (const float*)d_in[8];
    const float* a_dtb   = (const float*)d_in[9];
    const float* a_Alog  = (const float*)d_in[10];
    const float* a_D     = (const float*)d_in[11];
    const float* a_outw  = (const float*)d_in[12];
    const float* ln_a_g  = (const float*)d_in[13];
    const float* ln_a_b  = (const float*)d_in[14];
    const float* lin2_w  = (const float*)d_in[15];
    const float* lin2_b  = (const float*)d_in[16];
    const float* b_inw   = (const float*)d_in[17];
    const float* b_cw    = (const float*)d_in[18];
    const float* b_cb    = (const float*)d_in[19];
    const float* b_xpw   = (const float*)d_in[20];
    const float* b_dtw   = (const float*)d_in[21];
    const float* b_dtb   = (const float*)d_in[22];
    const float* b_Alog  = (const float*)d_in[23];
    const float* b_D     = (const float*)d_in[24];
    const float* b_outw  = (const float*)d_in[25];
    const float* ln_b_g  = (const float*)d_in[26];
    const float* ln_b_b  = (const float*)d_in[27];

    float* ws = (float*)d_ws;
    // workspace layout (floats), with lifetime-based reuse
    float* t    = ws + 0;                  // 6,291,456  (t -> yin_a -> y2/yin_b)
    float* br1  = ws + 6291456;            // 2,097,152
    float* xz   = ws + 8388608;            // 12,582,912 (xz_a -> xz_b)
    float* xc   = ws + 20971520;           // 6,291,456  (xc_a -> t2 -> xc_b)
    float* dtb_ = ws + 27262976;           // 6,291,456  (dt_a -> y_a -> dt_b | y_b)
    float* bc_a = ws + 33554432;           // 65,536
    float* bc_b = ws + 33619968;           // 65,536
    float* yin_a = t;
    float* y2    = t;
    float* yin_b = t + 2359296;
    float* y_a   = dtb_;
    float* t2    = xc;
    float* dt_b  = dtb_;
    float* xc_b  = xc;
    float* y_b   = dtb_ + 3145728;

    // ---- front end ----
    prep_kernel<<<24576, 256, 0, stream>>>(x1, x2, t);

    // ---- branch 1: silu(t @ lin1^T + b) ----
    gemm_wmma<true, true><<<dim3(1, 1024), 256, 0, stream>>>(t, lin1_w, lin1_b, br1,
                                                             MTOK, 128, DA);

    // ---- mamba a (d=384) ----
    gemm_wmma<false, false><<<dim3(6, 1024), 256, 0, stream>>>(t, a_inw, nullptr, xz,
                                                               MTOK, 768, DA);
    conv_kernel<<<24576, 256, 0, stream>>>(xz, a_cw, a_cb, xc, DA, 768, MTOK * DA);
    xproj_kernel<<<MTOK, 128, 0, stream>>>(xc, a_xpw, a_dtw, a_dtb, dtb_, bc_a, DA, 24);
    scan_kernel<<<6, 256, 0, stream>>>(xc, dtb_, bc_a, xz, a_Alog, a_D, yin_a,
                                       DA, 768, BATCH * DA);
    gemm_wmma<false, false><<<dim3(3, 1024), 256, 0, stream>>>(yin_a, a_outw, nullptr, y_a,
                                                               MTOK, DA, DA);
    ln_kernel<<<2048, 256, 0, stream>>>(y_a, ln_a_g, ln_a_b, t2, DA, MTOK);

    // ---- mid: silu(t2 @ lin2^T + b) ----
    gemm_wmma<true, true><<<dim3(1, 1024), 256, 0, stream>>>(t2, lin2_w, lin2_b, y2,
                                                             MTOK, 128, DA);

    // ---- mamba b (d=128) ----
    gemm_wmma<false, false><<<dim3(2, 1024), 256, 0, stream>>>(y2, b_inw, nullptr, xz,
                                                               MTOK, 256, DB);
    conv_kernel<<<8192, 256, 0, stream>>>(xz, b_cw, b_cb, xc_b, DB, 256, MTOK * DB);
    xproj_kernel<<<MTOK, 128, 0, stream>>>(xc_b, b_xpw, b_dtw, b_dtb, dt_b, bc_b, DB, 8);
    scan_kernel<<<2, 256, 0, stream>>>(xc_b, dt_b, bc_b, xz, b_Alog, b_D, yin_b,
                                       DB, 256, BATCH * DB);
    gemm_wmma<false, false><<<dim3(1, 1024), 256, 0, stream>>>(yin_b, b_outw, nullptr, y_b,
                                                               MTOK, DB, DB);

    // ---- final: LN_b + residual, NCHW store ----
    final_kernel<<<2048, 256, 0, stream>>>(y_b, ln_b_g, ln_b_b, br1, (float*)d_out);
}